// GraphAttentionLayer_31327491457337
// MI455X (gfx1250) — compile-verified
//
#include <hip/hip_runtime.h>

typedef __attribute__((ext_vector_type(2))) float v2f;
typedef __attribute__((ext_vector_type(8))) float v8f;

#define N_NODES   512
#define IN_F      512
#define OUT_F     512   // N_HEADS * N_HID
#define N_HEADS   8
#define N_HID     64
#define NEG_SLOPE 0.2f

// ---------------------------------------------------------------------------
// Kernel 1: dual projection GEMM  gs = x @ Ws^T, gt = x @ Wt^T  (fp32 WMMA)
// One wave per 16x16 output tile; A fragment shared between both products.
// fp32 WMMA fragment layout (ISA 7.12.2):
//   A 16x4 : lane l<16 -> row l, K={0,1} in v[0],v[1]; lane l+16 -> K={2,3}
//   B 4x16 : lane l<16 -> col l, K={0,1};             lane l+16 -> K={2,3}
//   C/D    : vgpr r, lane l<16 -> (M=r, N=l); lane l+16 -> (M=r+8, N=l-16)
// ---------------------------------------------------------------------------
__global__ __launch_bounds__(128) void gat_proj_gemm(
    const float* __restrict__ x,
    const float* __restrict__ Ws,
    const float* __restrict__ Wt,
    float* __restrict__ gs,
    float* __restrict__ gt)
{
    const int lane = threadIdx.x & 31;
    const int wave = threadIdx.x >> 5;
    const int tile = blockIdx.x * 4 + wave;     // 0..1023 (32x32 tiles)
    const int m0   = (tile >> 5) << 4;
    const int n0   = (tile & 31) << 4;
    const int lid  = lane & 15;
    const int half = lane >> 4;

    const float* arow  = x  + (m0 + lid) * IN_F + 2 * half;
    const float* bsrow = Ws + (n0 + lid) * IN_F + 2 * half;   // B = W^T: row n of W
    const float* btrow = Wt + (n0 + lid) * IN_F + 2 * half;

    v8f cs = {};
    v8f ct = {};
    for (int k0 = 0; k0 < IN_F; k0 += 4) {
        v2f a  = *reinterpret_cast<const v2f*>(arow  + k0);
        v2f bs = *reinterpret_cast<const v2f*>(bsrow + k0);
        v2f bt = *reinterpret_cast<const v2f*>(btrow + k0);
        cs = __builtin_amdgcn_wmma_f32_16x16x4_f32(false, a, false, bs,
                                                   (short)0, cs, false, false);
        ct = __builtin_amdgcn_wmma_f32_16x16x4_f32(false, a, false, bt,
                                                   (short)0, ct, false, false);
    }
#pragma unroll
    for (int r = 0; r < 8; ++r) {
        const int row = m0 + r + 8 * half;
        gs[row * OUT_F + n0 + lid] = cs[r];
        gt[row * OUT_F + n0 + lid] = ct[r];
    }
}

// ---------------------------------------------------------------------------
// Kernel 2: masked attention scores.
// sc[i,h,j] = sum_f leaky(gs[j,h,f] + gt[i,h,f]) * w[f]   (or -inf if adj==0)
// The |x| inside leaky prevents matmul factorization -> VALU loop over f.
// One block per (i,h); gt row + attn_w staged in LDS.
// ---------------------------------------------------------------------------
__global__ __launch_bounds__(256) void gat_scores(
    const float* __restrict__ gs,
    const float* __restrict__ gt,
    const float* __restrict__ attn_w,
    const int*   __restrict__ adj,
    float* __restrict__ sc)           // [N, H, N] (j contiguous)
{
    __shared__ float t_sh[N_HID];
    __shared__ float w_sh[N_HID];
    const int i = blockIdx.x >> 3;
    const int h = blockIdx.x & 7;

    if (threadIdx.x < N_HID) {
        t_sh[threadIdx.x] = gt[i * OUT_F + h * N_HID + threadIdx.x];
        w_sh[threadIdx.x] = attn_w[threadIdx.x];
    }
    __syncthreads();

    for (int j = threadIdx.x; j < N_NODES; j += 256) {
        const float* srow = gs + j * OUT_F + h * N_HID;
        float acc = 0.f;
#pragma unroll 8
        for (int f = 0; f < N_HID; ++f) {
            float v = srow[f] + t_sh[f];
            v = (v >= 0.f) ? v : NEG_SLOPE * v;   // LeakyReLU
            acc = fmaf(v, w_sh[f], acc);
        }
        sc[(i * N_HEADS + h) * N_NODES + j] =
            (adj[i * N_NODES + j] == 0) ? -__builtin_inff() : acc;
    }
}

// ---------------------------------------------------------------------------
// Kernel 3: softmax over j for each (i,h). Normalizes sc in place (kept as the
// A matrix for kernel 4) and writes attention to d_out in [i,j,h] layout.
// ---------------------------------------------------------------------------
__global__ __launch_bounds__(256) void gat_softmax(
    float* __restrict__ sc,           // [N, H, N] in/out
    float* __restrict__ attn_out)     // [N, N, H]
{
    __shared__ float red[256];
    const int i = blockIdx.x >> 3;
    const int h = blockIdx.x & 7;
    const int t = threadIdx.x;

    float* row = sc + (i * N_HEADS + h) * N_NODES;
    const float s0 = row[t];
    const float s1 = row[t + 256];

    red[t] = fmaxf(s0, s1);
    __syncthreads();
    for (int off = 128; off > 0; off >>= 1) {
        if (t < off) red[t] = fmaxf(red[t], red[t + off]);
        __syncthreads();
    }
    const float m = red[0];
    __syncthreads();

    const float e0 = expf(s0 - m);
    const float e1 = expf(s1 - m);
    red[t] = e0 + e1;
    __syncthreads();
    for (int off = 128; off > 0; off >>= 1) {
        if (t < off) red[t] += red[t + off];
        __syncthreads();
    }
    const float inv = 1.0f / red[0];

    const float a0 = e0 * inv;
    const float a1 = e1 * inv;
    row[t]       = a0;
    row[t + 256] = a1;
    attn_out[(i * N_NODES + t)       * N_HEADS + h] = a0;
    attn_out[(i * N_NODES + t + 256) * N_HEADS + h] = a1;
}

// ---------------------------------------------------------------------------
// Kernel 4: per-head output GEMM  out[i, h*64+f] = sum_j attn[i,h,j]*gt[j,h*64+f]
// One wave per 16x16 tile; 8 heads x (32 x 4) tiles = 1024 tiles. fp32 WMMA.
// ---------------------------------------------------------------------------
__global__ __launch_bounds__(128) void gat_out_gemm(
    const float* __restrict__ attn,   // [N, H, N] normalized
    const float* __restrict__ gt,     // [N, OUT_F]
    float* __restrict__ out)          // [N, OUT_F]
{
    const int lane = threadIdx.x & 31;
    const int wave = threadIdx.x >> 5;
    const int tile = blockIdx.x * 4 + wave;   // 0..1023
    const int head = tile >> 7;               // 128 tiles per head
    const int rem  = tile & 127;
    const int m0   = (rem >> 2) << 4;         // 32 row tiles (i)
    const int n0   = (rem & 3) << 4;          // 4 col tiles (f)
    const int lid  = lane & 15;
    const int half = lane >> 4;

    const float* arow = attn + ((m0 + lid) * N_HEADS + head) * N_NODES + 2 * half;
    const float* bcol = gt + head * N_HID + n0 + lid;   // column n of B, stride OUT_F

    v8f c = {};
    for (int k0 = 0; k0 < N_NODES; k0 += 4) {
        v2f a = *reinterpret_cast<const v2f*>(arow + k0);
        v2f b;
        b.x = bcol[(k0 + 2 * half)     * OUT_F];
        b.y = bcol[(k0 + 2 * half + 1) * OUT_F];
        c = __builtin_amdgcn_wmma_f32_16x16x4_f32(false, a, false, b,
                                                  (short)0, c, false, false);
    }
#pragma unroll
    for (int r = 0; r < 8; ++r) {
        out[(m0 + r + 8 * half) * OUT_F + head * N_HID + n0 + lid] = c[r];
    }
}

// ---------------------------------------------------------------------------
extern "C" void kernel_launch(void* const* d_in, const int* in_sizes, int n_in,
                              void* d_out, int out_size, void* d_ws, size_t ws_size,
                              hipStream_t stream) {
    const float* x   = (const float*)d_in[0];   // h [1,512,512]
    const float* Ws  = (const float*)d_in[1];   // W_source [512,512]
    const float* Wt  = (const float*)d_in[2];   // W_target [512,512]
    const float* aw  = (const float*)d_in[3];   // attn_w [64]
    const int*   adj = (const int*)d_in[4];     // adjacency [512,512,1]

    float* out      = (float*)d_out;                          // [512, 512]
    float* attn_out = (float*)d_out + N_NODES * OUT_F;        // [512, 512, 8]

    float* gs = (float*)d_ws;                  // [512, 512]   1 MB
    float* gt = gs + N_NODES * OUT_F;          // [512, 512]   1 MB
    float* sc = gt + N_NODES * OUT_F;          // [512, 8, 512] 8 MB

    gat_proj_gemm<<<256, 128, 0, stream>>>(x, Ws, Wt, gs, gt);
    gat_scores  <<<N_NODES * N_HEADS, 256, 0, stream>>>(gs, gt, aw, adj, sc);
    gat_softmax <<<N_NODES * N_HEADS, 256, 0, stream>>>(sc, attn_out);
    gat_out_gemm<<<256, 128, 0, stream>>>(sc, gt, out);
}